// HypergraphLayer_44178033606976
// MI455X (gfx1250) — compile-verified
//
#include <hip/hip_runtime.h>

typedef __attribute__((ext_vector_type(2))) float v2f;
typedef __attribute__((ext_vector_type(8))) float v8f;

#define LN_EPS 1e-5f

// ---------------------------------------------------------------------------
// Kernel 0: zero the aggregation workspace (d_ws is poisoned, not zeroed)
// ---------------------------------------------------------------------------
__global__ void zero_f32(float* __restrict__ p, long long n) {
  long long i = (long long)blockIdx.x * blockDim.x + threadIdx.x;
  const long long stride = (long long)gridDim.x * blockDim.x;
  for (; i < n; i += stride) p[i] = 0.0f;
}

// ---------------------------------------------------------------------------
// Kernel 1: per-edge leave-one-out message computation + scatter-add.
// Block = (D) threads, one channel per lane. grid = (E, B).
// Bandwidth/atomic bound; A is tiny (6) so fully unrolled in registers.
// ---------------------------------------------------------------------------
template <int AMAX>
__global__ void scatter_msgs(const float* __restrict__ nodef,
                             const int* __restrict__ edge_list,
                             const int* __restrict__ rel,
                             const float* __restrict__ pos_emb,
                             const float* __restrict__ rel_emb,
                             const float* __restrict__ alphaPtr,
                             float* __restrict__ agg,
                             int N, int Dn, int Acnt) {
  const int d = threadIdx.x;
  const int e = blockIdx.x;
  const int b = blockIdx.y;

  const float alpha = alphaPtr[0];
  const int relIdx = rel[e];
  // rel_emb row 0 is forced to all-ones in the reference
  const float relv = (relIdx == 0) ? 1.0f : rel_emb[(size_t)relIdx * Dn + d];

  int   idxs[AMAX];
  float s[AMAX];
#pragma unroll
  for (int a = 0; a < AMAX; ++a) {
    if (a < Acnt) {
      const int idx = edge_list[(size_t)e * Acnt + a];
      idxs[a] = idx;
      // node 0 features are zeroed; pos_emb row 0 is forced to all-ones
      const float g  = (idx == 0) ? 0.0f : nodef[((size_t)b * N + idx) * Dn + d];
      const float pe = (idx != 0) ? pos_emb[(size_t)(a + 1) * Dn + d] : 1.0f;
      s[a] = alpha * g + (1.0f - alpha) * pe;
    } else {
      idxs[a] = 0;
      s[a] = 1.0f;  // neutral for products
    }
  }

  float pre[AMAX], suf[AMAX];
  float p = 1.0f;
#pragma unroll
  for (int a = 0; a < AMAX; ++a) { pre[a] = p; p *= s[a]; }
  p = 1.0f;
#pragma unroll
  for (int a = AMAX - 1; a >= 0; --a) { suf[a] = p; p *= s[a]; }

#pragma unroll
  for (int a = 0; a < AMAX; ++a) {
    if (a < Acnt && idxs[a] != 0) {  // agg[:,0,:] is zeroed afterwards anyway
      atomicAdd(&agg[((size_t)b * N + idxs[a]) * Dn + d], pre[a] * suf[a] * relv);
    }
  }
}

// ---------------------------------------------------------------------------
// Kernel 2: fused  out = LN( concat(agg, nf) @ W + b ) via V_WMMA_F32_16X16X4_F32.
// 256-thread block = 8 waves; each wave owns one 16-row x 64-col output tile.
// W is staged into LDS K-pair interleaved: sW2[p][col] = {W[2p][col], W[2p+1][col]}
// so every B fragment is a single aligned ds_load_b64 straight into an even
// VGPR pair (no register shuffles), addressed as base + 16-bit immediate.
// A fragments stream from HBM with the node-0 zeroing folded into a
// multiplicative mask, keeping the hot loop branch-free.
// ---------------------------------------------------------------------------
__global__ void gemm_ln_wmma(const float* __restrict__ agg,
                             const float* __restrict__ nodef,
                             const float* __restrict__ W,
                             const float* __restrict__ bias,
                             const float* __restrict__ gamma,
                             const float* __restrict__ beta,
                             float* __restrict__ out,
                             int N, int numRows, int numTiles) {
  __shared__ float2 sW2[64 * 64];  // p = K-pair (0..63), col (0..63): 32 KB
  {
#pragma unroll
    for (int i = 0; i < 16; ++i) {
      const int idx = threadIdx.x + i * 256;   // 4096 entries / 256 threads
      const int p = idx >> 6;
      const int c = idx & 63;
      float2 v;
      v.x = W[(size_t)(2 * p) * 64 + c];
      v.y = W[(size_t)(2 * p + 1) * 64 + c];
      sW2[idx] = v;
    }
  }
  __syncthreads();

  const int wave = threadIdx.x >> 5;
  const int lane = threadIdx.x & 31;
  const int tile = blockIdx.x * 8 + wave;
  if (tile >= numTiles) return;  // wave-uniform: EXEC stays all-ones for WMMA

  const int laneN = lane & 15;   // M for A-frag, N-col for B/C/D frags
  const int h     = lane >> 4;   // half-wave selects K pair / row half

  int gr = tile * 16 + laneN;
  if (gr >= numRows) gr = numRows - 1;  // tail clamp (stores are guarded)
  const int bb = gr / N;
  const int nn = gr - bb * N;
  const float* aggRow = agg + (size_t)gr * 64 + 2 * h;
  const float* nfRow  = nodef + ((size_t)bb * N + nn) * 64 + 2 * h;
  // nf row 0 is zeroed in the reference: mask instead of branching so EXEC
  // stays uniform and loads pipeline freely.
  const float sel = (nn == 0) ? 0.0f : 1.0f;

  // this lane's K-pair index is p = 2k + h  ->  element index k*128 + h*64 + col
  const float2* wPtr = sW2 + h * 64 + laneN;

  v8f c0 = {}, c1 = {}, c2 = {}, c3 = {};

  // ---- first 64 of concat: agg ----
#pragma unroll
  for (int k = 0; k < 16; ++k) {
    const v2f a  = *(const v2f*)(aggRow + k * 4);
    const v2f b0 = *(const v2f*)(wPtr + k * 128);
    const v2f b1 = *(const v2f*)(wPtr + k * 128 + 16);
    const v2f b2 = *(const v2f*)(wPtr + k * 128 + 32);
    const v2f b3 = *(const v2f*)(wPtr + k * 128 + 48);
    c0 = __builtin_amdgcn_wmma_f32_16x16x4_f32(false, a, false, b0, (short)0, c0, false, false);
    c1 = __builtin_amdgcn_wmma_f32_16x16x4_f32(false, a, false, b1, (short)0, c1, false, false);
    c2 = __builtin_amdgcn_wmma_f32_16x16x4_f32(false, a, false, b2, (short)0, c2, false, false);
    c3 = __builtin_amdgcn_wmma_f32_16x16x4_f32(false, a, false, b3, (short)0, c3, false, false);
  }

  // ---- second 64 of concat: node features (masked for node 0) ----
#pragma unroll
  for (int k = 0; k < 16; ++k) {
    const v2f t = *(const v2f*)(nfRow + k * 4);
    v2f a; a.x = t.x * sel; a.y = t.y * sel;
    const v2f b0 = *(const v2f*)(wPtr + (k + 16) * 128);
    const v2f b1 = *(const v2f*)(wPtr + (k + 16) * 128 + 16);
    const v2f b2 = *(const v2f*)(wPtr + (k + 16) * 128 + 32);
    const v2f b3 = *(const v2f*)(wPtr + (k + 16) * 128 + 48);
    c0 = __builtin_amdgcn_wmma_f32_16x16x4_f32(false, a, false, b0, (short)0, c0, false, false);
    c1 = __builtin_amdgcn_wmma_f32_16x16x4_f32(false, a, false, b1, (short)0, c1, false, false);
    c2 = __builtin_amdgcn_wmma_f32_16x16x4_f32(false, a, false, b2, (short)0, c2, false, false);
    c3 = __builtin_amdgcn_wmma_f32_16x16x4_f32(false, a, false, b3, (short)0, c3, false, false);
  }

  // ---- fused bias + LayerNorm ----
  const float bi0 = bias[laneN],      bi1 = bias[16 + laneN];
  const float bi2 = bias[32 + laneN], bi3 = bias[48 + laneN];
  const float g0 = gamma[laneN],      g1 = gamma[16 + laneN];
  const float g2 = gamma[32 + laneN], g3 = gamma[48 + laneN];
  const float be0 = beta[laneN],      be1 = beta[16 + laneN];
  const float be2 = beta[32 + laneN], be3 = beta[48 + laneN];

  float sum[8], ssq[8];
#pragma unroll
  for (int j = 0; j < 8; ++j) {
    c0[j] += bi0; c1[j] += bi1; c2[j] += bi2; c3[j] += bi3;
    float sm = c0[j] + c1[j] + c2[j] + c3[j];
    float sq = c0[j] * c0[j] + c1[j] * c1[j] + c2[j] * c2[j] + c3[j] * c3[j];
    // reduce across the 16 lanes of this half-wave (xor<=8 stays in-half)
#pragma unroll
    for (int m = 1; m <= 8; m <<= 1) {
      sm += __shfl_xor(sm, m, 32);
      sq += __shfl_xor(sq, m, 32);
    }
    sum[j] = sm; ssq[j] = sq;
  }

#pragma unroll
  for (int j = 0; j < 8; ++j) {
    const int row = tile * 16 + 8 * h + j;
    if (row >= numRows) continue;
    const float mu  = sum[j] * (1.0f / 64.0f);
    const float var = ssq[j] * (1.0f / 64.0f) - mu * mu;
    const float inv = rsqrtf(var + LN_EPS);
    float* orow = out + (size_t)row * 64;
    orow[laneN]      = (c0[j] - mu) * inv * g0 + be0;
    orow[16 + laneN] = (c1[j] - mu) * inv * g1 + be1;
    orow[32 + laneN] = (c2[j] - mu) * inv * g2 + be2;
    orow[48 + laneN] = (c3[j] - mu) * inv * g3 + be3;
  }
}

// ---------------------------------------------------------------------------
extern "C" void kernel_launch(void* const* d_in, const int* in_sizes, int n_in,
                              void* d_out, int out_size, void* d_ws, size_t ws_size,
                              hipStream_t stream) {
  const float* node_features = (const float*)d_in[0];
  // d_in[1] = query (unused by the reference forward)
  const int*   edge_list = (const int*)d_in[2];
  const int*   rel       = (const int*)d_in[3];
  const float* pos_emb   = (const float*)d_in[4];
  const float* rel_emb   = (const float*)d_in[5];
  const float* W         = (const float*)d_in[6];
  const float* bvec      = (const float*)d_in[7];
  const float* gamma     = (const float*)d_in[8];
  const float* beta      = (const float*)d_in[9];
  const float* alphaPtr  = (const float*)d_in[10];

  const int E  = in_sizes[3];
  const int A  = in_sizes[2] / E;              // 6
  const int D  = in_sizes[4] / (A + 1);        // 64
  const int B  = in_sizes[1] / D;              // 2
  const int N  = in_sizes[0] / (B * D);        // 50000

  float* agg = (float*)d_ws;                   // B*N*D floats
  const long long aggN = (long long)B * N * D;

  // 1) zero the accumulator
  zero_f32<<<4096, 256, 0, stream>>>(agg, aggN);

  // 2) edge messages + scatter-add
  dim3 sg(E, B, 1);
  if (A <= 8) {
    scatter_msgs<8><<<sg, D, 0, stream>>>(node_features, edge_list, rel, pos_emb,
                                          rel_emb, alphaPtr, agg, N, D, A);
  } else {
    scatter_msgs<16><<<sg, D, 0, stream>>>(node_features, edge_list, rel, pos_emb,
                                           rel_emb, alphaPtr, agg, N, D, A);
  }

  // 3) fused WMMA GEMM (+bias) + LayerNorm
  const int numRows  = B * N;
  const int numTiles = (numRows + 15) / 16;
  const int blocks   = (numTiles + 7) / 8;
  gemm_ln_wmma<<<blocks, 256, 0, stream>>>(agg, node_features, W, bvec, gamma, beta,
                                           (float*)d_out, N, numRows, numTiles);
}